// DGCN_link_prediction_70420283785588
// MI455X (gfx1250) — compile-verified
//
#include <hip/hip_runtime.h>
#include <hip/hip_bf16.h>

// ---------------------------------------------------------------------------
// DGCN link prediction for MI455X (gfx1250, wave32).
//
// Pipeline (all on `stream`, all state recomputed every call):
//   1. deg_c = 1 + segment_sum(w_c, col_c)  (3 edge sets, f32 atomics)
//   2. dinv_c = rsqrt(deg_c)
//   3. norm_c[e] = dinv_c[row]*w*dinv_c[col]   (cached, reused by all 3 blocks)
//   4. H1 = A @ W^T  via V_WMMA_F32_16X16X4_F32  (16x64 tile per wave)
//   5. B1[i, c*64+f] = dinv_c[i]^2 * H1[i,f] + bias[f]   (self loop + bias)
//   6. scatter: B1[col, c*64+f] += norm_c[e] * H1[row,f]  (float4 gather + atomics)
//   7. relu(B1) ; repeat 4-7 for blocks 2,3
//   8. per-query wave: 384-dot with Wl rows, shuffle-reduce, softmax
// ---------------------------------------------------------------------------

typedef __attribute__((ext_vector_type(2))) float v2f;
typedef __attribute__((ext_vector_type(8))) float v8f;

#define ATOMIC_ADD_F32(p, v) \
    __hip_atomic_fetch_add((p), (v), __ATOMIC_RELAXED, __HIP_MEMORY_SCOPE_AGENT)

// ---------------------------------------------------------------- utilities
__global__ __launch_bounds__(256) void fill_ones_kernel(float* __restrict__ p, long n) {
    long i = (long)blockIdx.x * 256 + threadIdx.x;
    if (i < n) p[i] = 1.0f;
}

__global__ __launch_bounds__(256) void accum_deg_kernel(const int* __restrict__ cols,
                                                        const float* __restrict__ w,
                                                        float* __restrict__ deg, int E) {
    int e = blockIdx.x * 256 + threadIdx.x;
    if (e >= E) return;
    float wv = (w != nullptr) ? w[e] : 1.0f;
    ATOMIC_ADD_F32(&deg[cols[e]], wv);
}

__global__ __launch_bounds__(256) void deg_to_dinv_kernel(float* __restrict__ p, long n) {
    long i = (long)blockIdx.x * 256 + threadIdx.x;
    if (i >= n) return;
    float d = p[i];
    p[i] = (d > 0.0f) ? rsqrtf(d) : 0.0f;
}

__global__ __launch_bounds__(256) void edge_norm_kernel(const int* __restrict__ rows,
                                                        const int* __restrict__ cols,
                                                        const float* __restrict__ w,
                                                        const float* __restrict__ dinv,
                                                        float* __restrict__ norm, int E) {
    int e = blockIdx.x * 256 + threadIdx.x;
    if (e >= E) return;
    float wv = (w != nullptr) ? w[e] : 1.0f;
    norm[e] = dinv[rows[e]] * wv * dinv[cols[e]];
}

// ------------------------------------------------------- dense GEMM via WMMA
// Out[N,64] = A[N,KD] @ W[64,KD]^T   (f32 WMMA 16x16x4, one 16x64 tile / wave)
template <int KD>
__global__ __launch_bounds__(256) void gemm_wmma_kernel(const float* __restrict__ A,
                                                        const float* __restrict__ W,
                                                        float* __restrict__ Out,
                                                        int nrows) {
    constexpr int KP = KD + 2;            // pad -> conflict-free, 8B aligned ds loads
    __shared__ float wlds[64 * KP];

    const int tid = threadIdx.x;
    for (int idx = tid; idx < 64 * KD; idx += 256) {
        int nn = idx / KD;
        int kk = idx - nn * KD;
        wlds[nn * KP + kk] = W[idx];
    }
    __syncthreads();

    const int wave  = tid >> 5;
    const int lane  = tid & 31;
    const int mtile = blockIdx.x * 8 + wave;
    if (mtile * 16 >= nrows) return;      // whole-wave skip keeps EXEC all-1s for WMMA

    const int m  = lane & 15;             // A row within tile / B column
    const int kk = (lane >> 4) * 2;       // K-pair select (ISA 16x4 f32 A layout)
    const float* arow = A + (long)(mtile * 16 + m) * KD;

    v8f acc0 = {}, acc1 = {}, acc2 = {}, acc3 = {};
    for (int k0 = 0; k0 < KD; k0 += 4) {
        v2f a = *(const v2f*)(arow + k0 + kk);
        const float* bp = wlds + m * KP + k0 + kk;
        v2f b0 = *(const v2f*)(bp);
        v2f b1 = *(const v2f*)(bp + 16 * KP);
        v2f b2 = *(const v2f*)(bp + 32 * KP);
        v2f b3 = *(const v2f*)(bp + 48 * KP);
        acc0 = __builtin_amdgcn_wmma_f32_16x16x4_f32(false, a, false, b0, (short)0, acc0, false, false);
        acc1 = __builtin_amdgcn_wmma_f32_16x16x4_f32(false, a, false, b1, (short)0, acc1, false, false);
        acc2 = __builtin_amdgcn_wmma_f32_16x16x4_f32(false, a, false, b2, (short)0, acc2, false, false);
        acc3 = __builtin_amdgcn_wmma_f32_16x16x4_f32(false, a, false, b3, (short)0, acc3, false, false);
    }

    // C/D layout: VGPR v -> M = v (lanes 0-15) or v+8 (lanes 16-31), N = lane&15
    const int mh = (lane >> 4) * 8;
    float* ob = Out + (long)(mtile * 16) * 64 + m;
#pragma unroll
    for (int v = 0; v < 8; ++v) {
        long r = (long)(v + mh) * 64;
        ob[r + 0]  = acc0[v];
        ob[r + 16] = acc1[v];
        ob[r + 32] = acc2[v];
        ob[r + 48] = acc3[v];
    }
}

// --------------------------------------------- block init: self-loop + bias
__global__ __launch_bounds__(256) void block_init_kernel(const float* __restrict__ h,
                                                         const float* __restrict__ dinv0,
                                                         const float* __restrict__ dinv1,
                                                         const float* __restrict__ dinv2,
                                                         const float* __restrict__ bias,
                                                         float* __restrict__ out, int n) {
    long idx = (long)blockIdx.x * 256 + threadIdx.x;
    if (idx >= (long)n * 64) return;
    int i = (int)(idx >> 6);
    int f = (int)(idx & 63);
    float hv = h[idx];
    float bf = bias[f];
    float d0 = dinv0[i], d1 = dinv1[i], d2 = dinv2[i];
    float* o = out + (long)i * 192 + f;
    o[0]   = d0 * d0 * hv + bf;
    o[64]  = d1 * d1 * hv + bf;
    o[128] = d2 * d2 * hv + bf;
}

// ---------------------------------------------------- edge message scatter
// 16 lanes per edge, float4 gather of h[row], 4 f32 atomics into out[col]
__global__ __launch_bounds__(256) void scatter_kernel(const float* __restrict__ h,
                                                      const int* __restrict__ rows,
                                                      const int* __restrict__ cols,
                                                      const float* __restrict__ norm,
                                                      float* __restrict__ out,
                                                      int E, int colOff) {
    int t = blockIdx.x * 256 + threadIdx.x;
    int e = t >> 4;
    int g = t & 15;
    if (e >= E) return;
    int r = rows[e];
    int c = cols[e];
    float w = norm[e];
    const float4 v = *(const float4*)(h + (long)r * 64 + g * 4);
    float* o = out + (long)c * 192 + colOff + g * 4;
    ATOMIC_ADD_F32(o + 0, w * v.x);
    ATOMIC_ADD_F32(o + 1, w * v.y);
    ATOMIC_ADD_F32(o + 2, w * v.z);
    ATOMIC_ADD_F32(o + 3, w * v.w);
}

__global__ __launch_bounds__(256) void relu_kernel(float* __restrict__ p, long n) {
    long i = (long)blockIdx.x * 256 + threadIdx.x;
    if (i < n) p[i] = fmaxf(p[i], 0.0f);
}

// --------------------------------------------------- query head + softmax
__global__ __launch_bounds__(256) void head_kernel(const float* __restrict__ feat,
                                                   const int* __restrict__ q,
                                                   const float* __restrict__ Wl,
                                                   const float* __restrict__ bl,
                                                   float* __restrict__ out, int Q) {
    int wid  = (int)(((long)blockIdx.x * 256 + threadIdx.x) >> 5);
    int lane = threadIdx.x & 31;
    if (wid >= Q) return;
    int a = q[wid * 2 + 0];
    int b = q[wid * 2 + 1];
    float acc0 = 0.0f, acc1 = 0.0f;
    for (int j = lane; j < 384; j += 32) {
        float v = (j < 192) ? feat[(long)a * 192 + j] : feat[(long)b * 192 + (j - 192)];
        acc0 += v * Wl[j];
        acc1 += v * Wl[384 + j];
    }
#pragma unroll
    for (int off = 16; off > 0; off >>= 1) {
        acc0 += __shfl_xor(acc0, off, 32);
        acc1 += __shfl_xor(acc1, off, 32);
    }
    if (lane == 0) {
        float l0 = acc0 + bl[0];
        float l1 = acc1 + bl[1];
        float mx = fmaxf(l0, l1);
        float e0 = expf(l0 - mx);
        float e1 = expf(l1 - mx);
        float s  = e0 + e1;
        out[wid * 2 + 0] = e0 / s;
        out[wid * 2 + 1] = e1 / s;
    }
}

// ---------------------------------------------------------------------------
extern "C" void kernel_launch(void* const* d_in, const int* in_sizes, int n_in,
                              void* d_out, int out_size, void* d_ws, size_t ws_size,
                              hipStream_t stream) {
    const float* x    = (const float*)d_in[0];
    const int*   ei   = (const int*)d_in[1];
    const int*   ein  = (const int*)d_in[2];
    const int*   eout = (const int*)d_in[3];
    const int*   qe   = (const int*)d_in[4];
    const float* inw  = (const float*)d_in[5];
    const float* outw = (const float*)d_in[6];
    const float* W1   = (const float*)d_in[7];
    const float* W2   = (const float*)d_in[8];
    const float* W3   = (const float*)d_in[9];
    const float* b1   = (const float*)d_in[10];
    const float* b2   = (const float*)d_in[11];
    const float* b3   = (const float*)d_in[12];
    const float* Wl   = (const float*)d_in[13];
    const float* bl   = (const float*)d_in[14];
    float* out = (float*)d_out;

    const int N = in_sizes[0] / 128;   // 100000
    const int E = in_sizes[5];         // 1600000
    const int Q = in_sizes[4] / 2;     // 10000

    // workspace carve-up (floats): 3N dinv + 3E norms + N*64 H1 + N*192 B1
    float* ws    = (float*)d_ws;
    float* dinv0 = ws;
    float* dinv1 = dinv0 + N;
    float* dinv2 = dinv1 + N;
    float* norm0 = dinv2 + N;
    float* norm1 = norm0 + E;
    float* norm2 = norm1 + E;
    float* H1    = norm2 + E;
    float* B1    = H1 + (long)N * 64;

    const int* rows0 = ei;   const int* cols0 = ei + E;
    const int* rows1 = ein;  const int* cols1 = ein + E;
    const int* rows2 = eout; const int* cols2 = eout + E;

    const int eb = (E + 255) / 256;

    // 1) degrees (self loop folded in as deg=1 init)
    fill_ones_kernel<<<(3L * N + 255) / 256, 256, 0, stream>>>(dinv0, 3L * N);
    accum_deg_kernel<<<eb, 256, 0, stream>>>(cols0, nullptr, dinv0, E);
    accum_deg_kernel<<<eb, 256, 0, stream>>>(cols1, inw,     dinv1, E);
    accum_deg_kernel<<<eb, 256, 0, stream>>>(cols2, outw,    dinv2, E);
    deg_to_dinv_kernel<<<(3L * N + 255) / 256, 256, 0, stream>>>(dinv0, 3L * N);

    // 2) per-edge norms (reused by all three blocks)
    edge_norm_kernel<<<eb, 256, 0, stream>>>(rows0, cols0, nullptr, dinv0, norm0, E);
    edge_norm_kernel<<<eb, 256, 0, stream>>>(rows1, cols1, inw,     dinv1, norm1, E);
    edge_norm_kernel<<<eb, 256, 0, stream>>>(rows2, cols2, outw,    dinv2, norm2, E);

    const int gemmBlocks = (N / 16 + 7) / 8;
    const long nh  = (long)N * 64;
    const long nb  = (long)N * 192;
    const int ibl  = (int)((nh + 255) / 256);
    const int rbl  = (int)((nb + 255) / 256);
    const int sbl  = (E * 16 + 255) / 256;

    const float* Ws[3]    = {W1, W2, W3};
    const float* biases[3]= {b1, b2, b3};

    for (int blk = 0; blk < 3; ++blk) {
        if (blk == 0)
            gemm_wmma_kernel<128><<<gemmBlocks, 256, 0, stream>>>(x,  Ws[0], H1, N);
        else
            gemm_wmma_kernel<192><<<gemmBlocks, 256, 0, stream>>>(B1, Ws[blk], H1, N);

        block_init_kernel<<<ibl, 256, 0, stream>>>(H1, dinv0, dinv1, dinv2, biases[blk], B1, N);
        scatter_kernel<<<sbl, 256, 0, stream>>>(H1, rows0, cols0, norm0, B1, E, 0);
        scatter_kernel<<<sbl, 256, 0, stream>>>(H1, rows1, cols1, norm1, B1, E, 64);
        scatter_kernel<<<sbl, 256, 0, stream>>>(H1, rows2, cols2, norm2, B1, E, 128);
        relu_kernel<<<rbl, 256, 0, stream>>>(B1, nb);
    }

    head_kernel<<<(Q * 32 + 255) / 256, 256, 0, stream>>>(B1, qe, Wl, bl, out, Q);
}